// CrossCorrelLoss_54511724921027
// MI455X (gfx1250) — compile-verified
//
#include <hip/hip_runtime.h>

typedef float v2f __attribute__((ext_vector_type(2)));
typedef float v8f __attribute__((ext_vector_type(8)));

#define DFEAT 321
#define DTILE 21          // ceil(321/16)
#define NPAIRS 231        // DTILE*(DTILE+1)/2 lower-triangle tile pairs
#define KTOT 32768        // B*T
#define GELEMS (DFEAT*DFEAT)
#define NSB 128           // stats blocks per input (256 rows each)

// workspace layout (floats):
//   [0      .. 671 ]  mu   : [2][336]  (sel-major)
//   [672    .. 1343]  inv  : [2][336]
//   [1344   ..      ]  G   : [2][321*321]
//   OFF_SP            stats partials: [2][NSB][2][336]  ({sum,sumsq})
//   OFF_AP            absdiff partials [128]
#define OFF_G    1344
#define OFF_SP   (OFF_G + 2*GELEMS)
#define OFF_AP   (OFF_SP + 2*NSB*2*336)

#if defined(__has_builtin)
#if __has_builtin(__builtin_amdgcn_wmma_f32_16x16x4_f32)
#define HAVE_WMMA_F32 1
#endif
#endif

// ---------------- Kernel 1a: per-block column sum/sumsq partials ------------
__global__ __launch_bounds__(336) void stats_partial_kernel(
    const float* __restrict__ xf, const float* __restrict__ xr,
    float* __restrict__ ws) {
  const int t   = threadIdx.x;      // feature (0..335)
  const int blk = blockIdx.x;       // 0..NSB-1
  const int sel = blockIdx.y;       // 0 fake, 1 real
  const float* X = sel ? xr : xf;
  float s = 0.f, q = 0.f;
  if (t < DFEAT) {
    const int r0 = blk * (KTOT / NSB);
    const float* p = X + (size_t)r0 * DFEAT + t;
    #pragma unroll 4
    for (int r = 0; r < KTOT / NSB; ++r) {
      float v = p[0];
      s += v; q += v * v;
      p += DFEAT;
    }
  }
  float* out = ws + OFF_SP + ((size_t)(sel * NSB + blk) * 2) * 336;
  out[t]       = s;
  out[336 + t] = q;
}

// ---------------- Kernel 1b: finalize mu / inv-std --------------------------
__global__ __launch_bounds__(336) void stats_final_kernel(float* __restrict__ ws) {
  const int t   = threadIdx.x;
  const int sel = blockIdx.x;
  float s = 0.f, q = 0.f;
  const float* base = ws + OFF_SP + (size_t)sel * NSB * 2 * 336;
  for (int b = 0; b < NSB; ++b) {
    s += base[(size_t)b * 2 * 336 + t];
    q += base[(size_t)b * 2 * 336 + 336 + t];
  }
  float m = 0.f, iv = 0.f;
  if (t < DFEAT) {
    m = s / (float)KTOT;
    float var = (q - (float)KTOT * m * m) / (float)(KTOT - 1);
    iv = rsqrtf(var);
  }
  ws[sel * 336 + t]       = m;
  ws[672 + sel * 336 + t] = iv;
}

// ---------------- Kernel 2: SYRK tiles via V_WMMA_F32_16X16X4_F32 -----------
// One workgroup (8 waves) per lower-triangle 16x16 tile pair per input.
// Each wave owns K slice {wave*4 + 32*n}; accumulators reduced through LDS.
// Out-of-range features load from a CLAMPED valid address (finite garbage);
// polluted rows/cols land only in outputs with dr/ec >= 321, which the
// writeback bounds check discards -> the hot loop is fully branchless.
__global__ __launch_bounds__(256) void gram_wmma_kernel(
    const float* __restrict__ xf, const float* __restrict__ xr,
    float* __restrict__ ws) {
  const int sel = blockIdx.y;
  const float* X   = sel ? xr : xf;
  const float* mu  = ws + sel * 336;
  const float* inv = ws + 672 + sel * 336;
  float* G = ws + OFF_G + (size_t)sel * GELEMS;

  // decode lower-triangle tile pair: blockIdx.x -> (ti >= tj)
  int p = blockIdx.x;
  int ti = 0;
  while ((ti + 1) * (ti + 2) / 2 <= p) ++ti;
  const int tj = p - ti * (ti + 1) / 2;

  const int tid   = threadIdx.x;
  const int lane  = tid & 31;
  const int wave  = tid >> 5;
  const int m     = lane & 15;   // M (for A) / N (for B) within tile
  const int khalf = lane >> 4;   // 0: K={0,1}, 1: K={2,3}

  int da = ti * 16 + m;          // A feature (row of output)
  int eb = tj * 16 + m;          // B feature (col of output)
  da = (da < DFEAT) ? da : (DFEAT - 1);   // clamp: keep addresses valid
  eb = (eb < DFEAT) ? eb : (DFEAT - 1);

  const float* pA = X + (size_t)(wave * 4 + khalf * 2) * DFEAT + da;
  const float* pB = X + (size_t)(wave * 4 + khalf * 2) * DFEAT + eb;

  v8f c = {0.f, 0.f, 0.f, 0.f, 0.f, 0.f, 0.f, 0.f};

  #pragma unroll 8
  for (int k0 = 0; k0 < KTOT / 8; k0 += 4) {   // per-wave K slice, stride 32
    v2f a, b;
    a.x = pA[0];
    a.y = pA[DFEAT];
    b.x = pB[0];
    b.y = pB[DFEAT];
#if defined(HAVE_WMMA_F32)
    // D = A(16x4 f32) x B(4x16 f32) + C(16x16 f32)
    c = __builtin_amdgcn_wmma_f32_16x16x4_f32(
        /*neg_a=*/false, a, /*neg_b=*/false, b,
        /*c_mod=*/(short)0, c, /*reuse_a=*/false, /*reuse_b=*/false);
#else
    c[0] += a.x * b.x + a.y * b.y;   // compile-only fallback
#endif
    pA += 32 * DFEAT;
    pB += 32 * DFEAT;
  }

  // reduce the 8 per-wave accumulators through LDS (deterministic)
  __shared__ float red[8 * 256];
  #pragma unroll
  for (int v = 0; v < 8; ++v) red[wave * 256 + lane * 8 + v] = c[v];
  __syncthreads();

  // 256 threads -> 256 output elements of the 16x16 tile
  const int olane = tid & 31;
  const int ocomp = tid >> 5;
  float tot = 0.f;
  #pragma unroll
  for (int w = 0; w < 8; ++w) tot += red[w * 256 + olane * 8 + ocomp];

  // C/D layout: VGPR v, lanes 0-15 -> M=v ; lanes 16-31 -> M=v+8 ; N=lane&15
  const int M  = ocomp + (olane >> 4) * 8;
  const int N  = olane & 15;
  const int dr = ti * 16 + M;
  const int ec = tj * 16 + N;
  if (dr < DFEAT && ec < DFEAT) {
    float val = (tot - (float)KTOT * mu[dr] * mu[ec]) * inv[dr] * inv[ec]
                * (1.0f / (float)KTOT);
    G[(size_t)dr * DFEAT + ec] = val;
  }
}

// ---------------- Kernel 3a: tril |Gf - Gr| block partials ------------------
__global__ __launch_bounds__(256) void absdiff_partial_kernel(
    const float* __restrict__ ws, float* __restrict__ partial) {
  const float* Gf = ws + OFF_G;
  const float* Gr = Gf + GELEMS;
  float s = 0.f;
  for (int q = blockIdx.x * 256 + threadIdx.x; q < GELEMS;
       q += gridDim.x * 256) {
    int d = q / DFEAT;
    int e = q - d * DFEAT;
    if (e <= d) s += fabsf(Gf[q] - Gr[q]);
  }
  __shared__ float ss[256];
  ss[threadIdx.x] = s;
  __syncthreads();
  for (int off = 128; off > 0; off >>= 1) {
    if (threadIdx.x < off) ss[threadIdx.x] += ss[threadIdx.x + off];
    __syncthreads();
  }
  if (threadIdx.x == 0) partial[blockIdx.x] = ss[0];
}

// ---------------- Kernel 3b: final scalar -----------------------------------
__global__ __launch_bounds__(256) void final_reduce_kernel(
    const float* __restrict__ partial, int n, float* __restrict__ out) {
  float s = 0.f;
  for (int i = threadIdx.x; i < n; i += 256) s += partial[i];
  __shared__ float ss[256];
  ss[threadIdx.x] = s;
  __syncthreads();
  for (int off = 128; off > 0; off >>= 1) {
    if (threadIdx.x < off) ss[threadIdx.x] += ss[threadIdx.x + off];
    __syncthreads();
  }
  if (threadIdx.x == 0) out[0] = ss[0] / 10.0f;
}

extern "C" void kernel_launch(void* const* d_in, const int* in_sizes, int n_in,
                              void* d_out, int out_size, void* d_ws, size_t ws_size,
                              hipStream_t stream) {
  const float* xf = (const float*)d_in[0];
  const float* xr = (const float*)d_in[1];
  float* ws = (float*)d_ws;

  stats_partial_kernel<<<dim3(NSB, 2), 336, 0, stream>>>(xf, xr, ws);
  stats_final_kernel<<<2, 336, 0, stream>>>(ws);
  gram_wmma_kernel<<<dim3(NPAIRS, 2), 256, 0, stream>>>(xf, xr, ws);

  float* partial = ws + OFF_AP;
  absdiff_partial_kernel<<<128, 256, 0, stream>>>(ws, partial);
  final_reduce_kernel<<<1, 256, 0, stream>>>(partial, 128, (float*)d_out);
}